// EGATConv_89567247991616
// MI455X (gfx1250) — compile-verified
//
#include <hip/hip_runtime.h>
#include <math.h>

typedef __attribute__((ext_vector_type(2))) float v2f;
typedef __attribute__((ext_vector_type(8))) float v8f;

#define CCH 128   // out channels
#define EDIM 32   // edge feature dim

// ---------------- init: zero out-accumulator & denom, amax = -inf ----------------
__global__ void init_kernel(float* __restrict__ out, float* __restrict__ denom,
                            int* __restrict__ amaxI, int N, long long totalNC) {
    long long i = (long long)blockIdx.x * blockDim.x + threadIdx.x;
    long long stride = (long long)gridDim.x * blockDim.x;
    for (; i < totalNC; i += stride) {
        out[i] = 0.0f;
        if (i < N) { denom[i] = 0.0f; amaxI[i] = (int)0xFF800000u; } // -inf bits
    }
}

// ---------------- fold: v_edge = W_edge @ att_edge  ([32,128]@[128] -> [32]) -----
__global__ void fold_edge_kernel(const float* __restrict__ W_edge,
                                 const float* __restrict__ att_edge,
                                 float* __restrict__ v_edge) {
    int k = threadIdx.x;                 // 32 threads
    float s = 0.0f;
    for (int c = 0; c < CCH; ++c) s += W_edge[k * CCH + c] * att_edge[c];
    v_edge[k] = s;
}

// ---------------- h = x@W (fp32 WMMA 16x16x4) + fused a_src/a_dst ----------------
// block = 256 threads = 8 waves; each wave owns a 16-row strip, all 8 column tiles.
__global__ __launch_bounds__(256)
void gemm_attn_kernel(const float* __restrict__ x, const float* __restrict__ W,
                      const float* __restrict__ att_src, const float* __restrict__ att_dst,
                      float* __restrict__ h, float* __restrict__ a_src,
                      float* __restrict__ a_dst, int N) {
    // Pre-swizzled B fragments for 16 k-steps x 8 N-tiles x 32 lanes (32 KB LDS)
    __shared__ v2f wfrag[16][8][32];

    const int tid  = threadIdx.x;
    const int wv   = tid >> 5;
    const int lane = tid & 31;
    const int nn   = lane & 15;
    const int row0 = blockIdx.x * 128 + wv * 16;

    // A fragment addressing: lanes 0-15 hold K={0,1}, lanes 16-31 hold K={2,3}
    const int mA     = row0 + nn;
    const int mClamp = (mA < N) ? mA : (N - 1);
    const int kOff   = (lane < 16) ? 0 : 2;
    const float* xrow = x + (long long)mClamp * CCH + kOff;

    v8f acc[8];
#pragma unroll
    for (int t = 0; t < 8; ++t)
#pragma unroll
        for (int j = 0; j < 8; ++j) acc[t][j] = 0.0f;

    for (int half = 0; half < 2; ++half) {
        __syncthreads();
        // stage B fragments: wfrag[s][t][lane] = {W[4S+k0][16t+n], W[4S+k0+1][16t+n]}
        for (int i = tid; i < 16 * 8 * 32; i += 256) {
            int s  = i >> 8;
            int t  = (i >> 5) & 7;
            int ln = i & 31;
            int S  = 16 * half + s;
            int k0 = 4 * S + ((ln < 16) ? 0 : 2);
            int cn = 16 * t + (ln & 15);
            v2f w2;
            w2[0] = W[k0 * CCH + cn];
            w2[1] = W[(k0 + 1) * CCH + cn];
            wfrag[s][t][ln] = w2;
        }
        __syncthreads();

#pragma unroll 4
        for (int s = 0; s < 16; ++s) {
            int S = 16 * half + s;
            v2f a = *reinterpret_cast<const v2f*>(xrow + 4 * S);
#pragma unroll
            for (int t = 0; t < 8; ++t) {
                acc[t] = __builtin_amdgcn_wmma_f32_16x16x4_f32(
                    false, a, false, wfrag[s][t][lane], (short)0, acc[t], false, false);
            }
        }
    }

    // C/D layout: VGPR v -> M = v (lanes 0-15) or v+8 (lanes 16-31); N = lane&15
    const int mBase = row0 + ((lane < 16) ? 0 : 8);
#pragma unroll
    for (int v = 0; v < 8; ++v) {
        int m = mBase + v;
        if (m < N) {
#pragma unroll
            for (int t = 0; t < 8; ++t)
                h[(long long)m * CCH + 16 * t + nn] = acc[t][v];
        }
    }

    // fused attention logits: a_src[m] = h[m,:] . att_src  (16-lane reduction)
    float as8[8], ad8[8];
#pragma unroll
    for (int t = 0; t < 8; ++t) {
        as8[t] = att_src[16 * t + nn];
        ad8[t] = att_dst[16 * t + nn];
    }
#pragma unroll
    for (int v = 0; v < 8; ++v) {
        float ps = 0.0f, pd = 0.0f;
#pragma unroll
        for (int t = 0; t < 8; ++t) {
            float cv = acc[t][v];
            ps += cv * as8[t];
            pd += cv * ad8[t];
        }
#pragma unroll
        for (int o = 1; o < 16; o <<= 1) {
            ps += __shfl_xor(ps, o, 32);
            pd += __shfl_xor(pd, o, 32);
        }
        int m = mBase + v;
        if (nn == 0 && m < N) { a_src[m] = ps; a_dst[m] = pd; }
    }
}

// ------------- per-edge logits + leaky-relu + segment max (8 lanes/edge) ---------
__global__ __launch_bounds__(256)
void edge_logit_kernel(const long long* __restrict__ ei, const float* __restrict__ edge_attr,
                       const float* __restrict__ v_edge, const float* __restrict__ a_src,
                       const float* __restrict__ a_dst, float* __restrict__ alpha,
                       int* __restrict__ amaxI, int E) {
    int g   = (int)((blockIdx.x * blockDim.x + threadIdx.x) >> 3);
    int sub = threadIdx.x & 7;
    if (g >= E) return;

    if (g + 1024 < E)  // software lookahead -> global_prefetch_b8
        __builtin_prefetch(edge_attr + (long long)(g + 1024) * EDIM + sub * 4, 0, 0);

    float4 q  = *reinterpret_cast<const float4*>(edge_attr + (long long)g * EDIM + sub * 4);
    float4 ve = *reinterpret_cast<const float4*>(v_edge + sub * 4);
    float p = q.x * ve.x + q.y * ve.y + q.z * ve.z + q.w * ve.w;
    p += __shfl_xor(p, 1, 32);
    p += __shfl_xor(p, 2, 32);
    p += __shfl_xor(p, 4, 32);

    if (sub == 0) {
        long long sI = ei[g];
        long long dI = ei[(long long)E + g];
        float al = a_src[sI] + a_dst[dI] + p;
        al = (al > 0.0f) ? al : 0.2f * al;          // leaky_relu(0.2)
        alpha[g] = al;
        // float-ordered max via native integer atomics (no CAS loop)
        if (al >= 0.0f)
            atomicMax((int*)&amaxI[dI], __float_as_int(al));
        else
            atomicMin((unsigned int*)&amaxI[dI], __float_as_uint(al));
    }
}

// -------- scatter: out[dst] += exp(alpha - amax[dst]) * h[src]; denom += ex ------
__global__ __launch_bounds__(256)
void scatter_kernel(const long long* __restrict__ ei, const float* __restrict__ alpha,
                    const int* __restrict__ amaxI, const float* __restrict__ h,
                    float* __restrict__ denom, float* __restrict__ out, int E) {
    int e    = (int)((blockIdx.x * blockDim.x + threadIdx.x) >> 5);
    int lane = threadIdx.x & 31;
    if (e >= E) return;

    long long sI = ei[e];
    long long dI = ei[(long long)E + e];
    float am = __int_as_float(amaxI[dI]);
    float ex = __expf(alpha[e] - am);

    if (lane == 0) atomicAdd(&denom[dI], ex);

    const float* hs = h + (long long)sI * CCH;
    float* od = out + (long long)dI * CCH;
#pragma unroll
    for (int i = 0; i < 4; ++i) {
        int c = lane + 32 * i;
        atomicAdd(&od[c], ex * hs[c]);   // global_atomic_add_f32, no-return
    }
}

// ---------------- finalize: out = acc/(denom+eps) + bias -------------------------
__global__ void finalize_kernel(float* __restrict__ out, const float* __restrict__ denom,
                                const float* __restrict__ bias, long long totalNC) {
    long long i = (long long)blockIdx.x * blockDim.x + threadIdx.x;
    long long stride = (long long)gridDim.x * blockDim.x;
    for (; i < totalNC; i += stride) {
        int n = (int)(i >> 7);
        int c = (int)(i & 127);
        out[i] = out[i] / (denom[n] + 1e-16f) + bias[c];
    }
}

extern "C" void kernel_launch(void* const* d_in, const int* in_sizes, int n_in,
                              void* d_out, int out_size, void* d_ws, size_t ws_size,
                              hipStream_t stream) {
    (void)n_in; (void)out_size; (void)ws_size;
    const float*     x        = (const float*)d_in[0];
    const long long* ei       = (const long long*)d_in[1];   // int64 [2,E]
    const float*     edge_attr= (const float*)d_in[2];
    const float*     W        = (const float*)d_in[3];
    const float*     W_edge   = (const float*)d_in[4];
    const float*     att_src  = (const float*)d_in[5];
    const float*     att_dst  = (const float*)d_in[6];
    const float*     att_edge = (const float*)d_in[7];
    const float*     bias     = (const float*)d_in[8];

    const int N = in_sizes[0] / CCH;     // 40000
    const int E = in_sizes[2] / EDIM;    // 640000
    float* out = (float*)d_out;

    // workspace layout (floats)
    float* ws     = (float*)d_ws;
    float* h      = ws;                          // N*C
    float* a_src  = h + (size_t)N * CCH;         // N
    float* a_dst  = a_src + N;                   // N
    float* denom  = a_dst + N;                   // N
    int*   amaxI  = (int*)(denom + N);           // N
    float* v_edge = (float*)(amaxI + N);         // 32 (16B-aligned)
    float* alpha  = v_edge + 32;                 // E

    const long long totalNC = (long long)N * CCH;

    init_kernel<<<1024, 256, 0, stream>>>(out, denom, amaxI, N, totalNC);
    fold_edge_kernel<<<1, 32, 0, stream>>>(W_edge, att_edge, v_edge);

    int gB = (N + 127) / 128;
    gemm_attn_kernel<<<gB, 256, 0, stream>>>(x, W, att_src, att_dst, h, a_src, a_dst, N);

    int gC = (int)(((long long)E * 8 + 255) / 256);
    edge_logit_kernel<<<gC, 256, 0, stream>>>(ei, edge_attr, v_edge, a_src, a_dst,
                                              alpha, amaxI, E);

    int gD = (int)(((long long)E * 32 + 255) / 256);
    scatter_kernel<<<gD, 256, 0, stream>>>(ei, alpha, amaxI, h, denom, out, E);

    finalize_kernel<<<2048, 256, 0, stream>>>(out, denom, bias, totalNC);
}